// Quanvolution_45294725104253
// MI455X (gfx1250) — compile-verified
//
#include <hip/hip_runtime.h>

typedef __attribute__((ext_vector_type(2))) float v2f;
typedef __attribute__((ext_vector_type(8))) float v8f;

// ---------------------------------------------------------------------------
// Kernel 1: build the fixed 16x16 layer-circuit matrix M (row-major).
// M[i][j] = (layer-circuit applied to basis vector e_j)[i].
// State index i = b0*8 + b1*4 + b2*2 + b3, where b_q is qubit q's bit.
// One thread per basis vector (column).
// ---------------------------------------------------------------------------
__global__ void build_circuit_matrix(const float* __restrict__ rp, int n_layers,
                                     float* __restrict__ M) {
  const int j = threadIdx.x;
  if (j >= 16) return;
  float st[16];
#pragma unroll
  for (int i = 0; i < 16; ++i) st[i] = (i == j) ? 1.0f : 0.0f;

  for (int l = 0; l < n_layers; ++l) {
    // RY(theta) on each qubit
#pragma unroll
    for (int q = 0; q < 4; ++q) {
      const float half = rp[l * 4 + q] * 0.5f;
      float s, c;
      __sincosf(half, &s, &c);
      const int m = 8 >> q;
#pragma unroll
      for (int i = 0; i < 16; ++i) {
        if ((i & m) == 0) {
          const float a0 = st[i], a1 = st[i | m];
          st[i]     = c * a0 - s * a1;
          st[i | m] = s * a0 + c * a1;
        }
      }
    }
    // CNOT ring (0,1),(1,2),(2,3),(3,0): swap target bit where ctrl bit == 1
    const int ctl[4] = {0, 1, 2, 3};
    const int tgt[4] = {1, 2, 3, 0};
#pragma unroll
    for (int g = 0; g < 4; ++g) {
      const int mc = 8 >> ctl[g];
      const int mt = 8 >> tgt[g];
#pragma unroll
      for (int i = 0; i < 16; ++i) {
        if ((i & mc) && !(i & mt)) {
          const float t = st[i];
          st[i] = st[i | mt];
          st[i | mt] = t;
        }
      }
    }
  }
#pragma unroll
  for (int i = 0; i < 16; ++i) M[i * 16 + j] = st[i];  // column j
}

// ---------------------------------------------------------------------------
// Kernel 2: one wave per tile of 16 consecutive patches.
// GEMM orientation: D[i][p] = sum_j M[i][j] * enc[p][j]  (A = M, B = enc^T).
// f32 WMMA per-lane K mapping (chunk c, vgpr r, half h): K = 4c + 2h + r,
// identical for A and B operands; A row = B col = lane&15.
// D layout: lane (h,n) holds final[state v+8h][patch n] in acc[v], v=0..7.
// ---------------------------------------------------------------------------
__global__ __launch_bounds__(256) void quanv_main(const float* __restrict__ x,
                                                  const float* __restrict__ Mws,
                                                  float* __restrict__ out) {
  const int lane = threadIdx.x & 31;
  const int wid  = threadIdx.x >> 5;
  const int h    = lane >> 4;   // half-wave
  const int n    = lane & 15;   // GEMM row (A) / col (B,D)
  const int tile = blockIdx.x * 8 + wid;   // 65536 tiles total

  // --- A operand: rows of M. Lane needs M[n][4c + 2h + {0,1}], c = 0..3.
  const float* mrow = Mws + n * 16 + 2 * h;
  v2f a[4];
#pragma unroll
  for (int c = 0; c < 4; ++c) {
    a[c].x = mrow[4 * c + 0];
    a[c].y = mrow[4 * c + 1];
  }

  // --- This lane's patch (lanes n and n+16 share a patch).
  const int p  = tile * 16 + n;            // patch id, row-major over 1024x1024
  const int pr = p >> 10;                  // patch row
  const int pc = p & 1023;                 // patch col
  const float* src = x + (size_t)(2 * pr) * 2048 + 2 * pc;
  const float2 top = *(const float2*)src;            // phi0, phi1
  const float2 bot = *(const float2*)(src + 2048);   // phi2, phi3

  // --- Encoding factors: f_q(0) = cos(pi*phi/2), f_q(1) = sin(pi*phi/2)
  const float HPI = 1.57079632679489662f;
  float s0, c0, s1, c1, s2, c2, s3, c3;
  __sincosf(top.x * HPI, &s0, &c0);
  __sincosf(top.y * HPI, &s1, &c1);
  __sincosf(bot.x * HPI, &s2, &c2);
  __sincosf(bot.y * HPI, &s3, &c3);

  // B operand: enc[k] with k = 4c + 2h + r -> bits b0=c>>1, b1=c&1, b2=h, b3=r
  const float g2 = h ? s2 : c2;
  const float q01[4] = {c0 * c1, c0 * s1, s0 * c1, s0 * s1};
  v2f b[4];
#pragma unroll
  for (int c = 0; c < 4; ++c) {
    const float base = q01[c] * g2;
    b[c].x = base * c3;   // r = 0 -> b3 = 0 -> cos
    b[c].y = base * s3;   // r = 1 -> b3 = 1 -> sin
  }

  // --- Chained WMMA: 16x16 = M(16x16) @ enc^T(16x16), K split into 4x(K=4)
  v8f acc = {};
  acc = __builtin_amdgcn_wmma_f32_16x16x4_f32(false, a[0], false, b[0],
                                              (short)0, acc, false, false);
  acc = __builtin_amdgcn_wmma_f32_16x16x4_f32(false, a[1], false, b[1],
                                              (short)0, acc, false, false);
  acc = __builtin_amdgcn_wmma_f32_16x16x4_f32(false, a[2], false, b[2],
                                              (short)0, acc, false, false);
  acc = __builtin_amdgcn_wmma_f32_16x16x4_f32(false, a[3], false, b[3],
                                              (short)0, acc, false, false);

  // --- Z expectations. Lane holds states i = v + 8h of its own patch.
  float pv[8];
#pragma unroll
  for (int v = 0; v < 8; ++v) pv[v] = acc[v] * acc[v];

  const float t0 = pv[0] + pv[1], t1 = pv[2] + pv[3];
  const float t2 = pv[4] + pv[5], t3 = pv[6] + pv[7];
  const float sum = (t0 + t1) + (t2 + t3);
  float e0 = h ? -sum : sum;                      // sign (-1)^b0, b0 = h
  float e1 = (t0 + t1) - (t2 + t3);               // sign (-1)^(v>>2)
  float e2 = (t0 - t1) + (t2 - t3);               // sign (-1)^((v>>1)&1)
  float e3 = (pv[0] - pv[1]) + (pv[2] - pv[3])    // sign (-1)^(v&1)
           + (pv[4] - pv[5]) + (pv[6] - pv[7]);

  // Combine the two half-waves (states 0..7 with states 8..15)
  e0 += __shfl_xor(e0, 16, 32);
  e1 += __shfl_xor(e1, 16, 32);
  e2 += __shfl_xor(e2, 16, 32);
  e3 += __shfl_xor(e3, 16, 32);

  if (h == 0) {
    float4 o;
    o.x = e0; o.y = e1; o.z = e2; o.w = e3;
    ((float4*)out)[p] = o;   // out[pr][pc][0..3]
  }
}

// ---------------------------------------------------------------------------
extern "C" void kernel_launch(void* const* d_in, const int* in_sizes, int n_in,
                              void* d_out, int out_size, void* d_ws, size_t ws_size,
                              hipStream_t stream) {
  const float* x  = (const float*)d_in[0];        // (2048,2048,1) f32
  const float* rp = (const float*)d_in[1];        // (n_layers,4) f32
  const int n_layers = in_sizes[1] / 4;
  float* Mws = (float*)d_ws;                      // 256 floats scratch

  build_circuit_matrix<<<1, 16, 0, stream>>>(rp, n_layers, Mws);

  // 1024*1024 patches / 16 per tile = 65536 tiles; 8 waves per block.
  quanv_main<<<8192, 256, 0, stream>>>(x, Mws, (float*)d_out);
}